// coherence_cropped_31782757991045
// MI455X (gfx1250) — compile-verified
//
#include <hip/hip_runtime.h>
#include <stdint.h>

// Problem constants (from the reference)
#define T_DIM   32
#define C_DIM   64
#define S_DIM   2000
#define SL_DIM  4
#define L_WIN   500
#define TSTRIDE 100
#define W_DIM   16          // (2000-500)/100 + 1

// K-chunking through LDS
#define KC      24          // samples per LDS chunk
#define PITCH   25          // odd pitch -> conflict-free fragment reads
#define NCHUNK  21          // 20*24 + 20 = 500
#define BUFBYTES (256 * PITCH * 4)

typedef __attribute__((ext_vector_type(2))) float v2f;
typedef __attribute__((ext_vector_type(8))) float v8f;

__global__ __launch_bounds__(512) void coherence_corr_kernel(
    const float* __restrict__ x, float* __restrict__ out) {
  // LDS: double-buffered K-chunk, rows = c*4+s (256 rows), cols = k (pitch 25)
  __shared__ float Xs[2][256 * PITCH];   // 51.2 KB
  __shared__ float smean[256];           // per (c,s): mean
  __shared__ float sinvd[256];           // per (c,s): raw sum(x^2), later 1/||xm||

  const int tid  = threadIdx.x;
  const int w    = blockIdx.x;           // 0..15
  const int t    = blockIdx.y;           // 0..31
  const int lane = tid & 31;
  const int L    = lane & 15;
  const int h    = lane >> 4;            // lane-half selects K pair / M+8
  const int wv   = tid >> 5;             // 0..15 waves
  const int s    = wv >> 2;              // 0..3 : SL slot this wave computes
  const int m    = wv & 3;               // 0..3 : row-tile this wave computes

  // base of x[t][0][w*100][0]; channel stride = 8000 floats
  const float* xb = x + (size_t)t * (C_DIM * S_DIM * SL_DIM)
                      + (size_t)(w * TSTRIDE) * SL_DIM;

  // ---------- pass 1: per-(c,s) sum and sum-of-squares over the window ----------
  if (tid < 256) { smean[tid] = 0.0f; sinvd[tid] = 0.0f; }
  __syncthreads();
  if (tid < 256) {
    const int c = tid >> 2, q = tid & 3;                 // quarter-window each
    const float* p = xb + c * (S_DIM * SL_DIM) + q * 125 * SL_DIM;
    float s0=0.f,s1=0.f,s2=0.f,s3=0.f,q0=0.f,q1=0.f,q2=0.f,q3=0.f;
    #pragma unroll 5
    for (int i = 0; i < 125; ++i) {
      const float4 v = *reinterpret_cast<const float4*>(p + i * 4);
      s0 += v.x; s1 += v.y; s2 += v.z; s3 += v.w;
      q0 += v.x*v.x; q1 += v.y*v.y; q2 += v.z*v.z; q3 += v.w*v.w;
    }
    atomicAdd(&smean[c*4+0], s0); atomicAdd(&sinvd[c*4+0], q0);
    atomicAdd(&smean[c*4+1], s1); atomicAdd(&sinvd[c*4+1], q1);
    atomicAdd(&smean[c*4+2], s2); atomicAdd(&sinvd[c*4+2], q2);
    atomicAdd(&smean[c*4+3], s3); atomicAdd(&sinvd[c*4+3], q3);
  }
  __syncthreads();
  if (tid < 256) {
    const float mu  = smean[tid] * (1.0f / (float)L_WIN);
    const float var = sinvd[tid] - (float)L_WIN * mu * mu;   // sum((x-mu)^2)
    smean[tid] = mu;
    sinvd[tid] = rsqrtf(var);
  }

  // ---------- async copy setup (computed once) ----------
  // Wave wv copies channels cw..cw+3; each issue: 32 lanes x 4B = 128B contiguous
  // global, LDS scatter row (c*4 + s), col k  -> 32 distinct banks.
  const int cw = wv * 4;
  const unsigned lbase =
      (unsigned)(uintptr_t)(&Xs[0][0]) +
      4u * (unsigned)((cw * 4 + (lane & 3)) * PITCH + (lane >> 2));
  const float* gbase = xb + cw * (S_DIM * SL_DIM) + lane;

  // GLOBAL_LOAD_ASYNC_TO_LDS_B32: IOFFSET is added to BOTH the global and the
  // LDS address, so bias the LDS VGPR by (Cl - Cg) and carry Cg as immediate.
  auto load_chunk_async = [&](int buf, int k0, bool partial) {
    const unsigned long long ga =
        (unsigned long long)(uintptr_t)(gbase + k0 * SL_DIM);
    const unsigned lds0 = lbase + (buf ? (unsigned)BUFBYTES : 0u);
    #pragma unroll
    for (int i = 0; i < 4; ++i) {           // channel cw+i
      #pragma unroll
      for (int g = 0; g < 3; ++g) {         // k-range g*8 .. g*8+7
        const int Cg = (i * (S_DIM * SL_DIM) + g * 32) * 4;   // global byte disp
        const int Cl = (i * 4 * PITCH + g * 8) * 4;           // lds byte disp
        const unsigned lv = lds0 + (unsigned)(Cl - Cg);
        if (!partial || g < 2) {
          asm volatile("global_load_async_to_lds_b32 %0, %1, off offset:%2"
                       :: "v"(lv), "v"(ga), "i"(Cg) : "memory");
        } else if (lane < 16) {             // kc==20: only k=16..19 of g==2
          asm volatile("global_load_async_to_lds_b32 %0, %1, off offset:%2"
                       :: "v"(lv), "v"(ga), "i"(Cg) : "memory");
        }
      }
    }
  };

  // ---------- pass 2: Gram matrix via V_WMMA_F32_16X16X4_F32 ----------
  v8f acc[4] = {};   // this wave's 4 tiles: (m, n=0..3) for slot s

  // A fragment row for this lane: channel 16m+L, slot s
  const int rowA  = (64 * m + 4 * L + s) * PITCH;
  // B fragment base row (n added as n*64*PITCH): channel 16n+L, slot s
  const int rowB0 = (4 * L + s) * PITCH;

  load_chunk_async(0, 0, false);
  asm volatile("s_wait_asynccnt 0" ::: "memory");
  __syncthreads();

  for (int chunk = 0; chunk < NCHUNK; ++chunk) {
    const int kc = (chunk == NCHUNK - 1) ? 20 : KC;
    if (chunk + 1 < NCHUNK) {
      load_chunk_async((chunk + 1) & 1, (chunk + 1) * KC,
                       (chunk + 1) == NCHUNK - 1);
    }
    const float* Xb = Xs[chunk & 1];
    const int ksteps = kc >> 2;           // 6 or 5 K-steps of 4
    for (int j = 0; j < ksteps; ++j) {
      const int k = 4 * j + 2 * h;        // lanes 0-15: K={k0,k0+1}; 16-31: {k0+2,k0+3}
      // batch ALL fragment loads before the WMMAs
      v2f a, b[4];
      a.x = Xb[rowA + k];
      a.y = Xb[rowA + k + 1];
      #pragma unroll
      for (int n = 0; n < 4; ++n) {
        b[n].x = Xb[rowB0 + n * (64 * PITCH) + k];
        b[n].y = Xb[rowB0 + n * (64 * PITCH) + k + 1];
      }
      #pragma unroll
      for (int n = 0; n < 4; ++n) {
        acc[n] = __builtin_amdgcn_wmma_f32_16x16x4_f32(
            /*neg_a=*/false, a, /*neg_b=*/false, b[n],
            /*c_mod=*/(short)0, acc[n], /*reuse_a=*/false, /*reuse_b=*/false);
      }
    }
    // next chunk's DMA must be in LDS before anyone crosses this barrier
    asm volatile("s_wait_asynccnt 0" ::: "memory");
    __syncthreads();
  }

  // ---------- finalize: cov = G - L*mu_c*mu_d ; corr = cov*inv_c*inv_d ; clip ----------
  #pragma unroll
  for (int n = 0; n < 4; ++n) {
    const int d      = 16 * n + L;        // output column this lane holds
    const float mu_d  = smean[d * 4 + s];
    const float inv_d = sinvd[d * 4 + s];
    #pragma unroll
    for (int r = 0; r < 8; ++r) {
      const int cM = 16 * m + r + 8 * h;  // output row for acc VGPR r
      const float mu_c  = smean[cM * 4 + s];
      const float inv_c = sinvd[cM * 4 + s];
      const float cov   = acc[n][r] - (float)L_WIN * mu_c * mu_d;
      float corr = cov * inv_c * inv_d;
      corr = fminf(1.0f, fmaxf(-1.0f, corr));
      out[(((size_t)t * C_DIM + cM) * C_DIM + d) * (SL_DIM * W_DIM) + (s * W_DIM + w)] = corr;
    }
  }
}

extern "C" void kernel_launch(void* const* d_in, const int* in_sizes, int n_in,
                              void* d_out, int out_size, void* d_ws, size_t ws_size,
                              hipStream_t stream) {
  (void)in_sizes; (void)n_in; (void)out_size; (void)d_ws; (void)ws_size;
  const float* x = (const float*)d_in[0];
  float* out = (float*)d_out;
  dim3 grid(W_DIM, T_DIM);   // one workgroup per (w, t)
  coherence_corr_kernel<<<grid, 512, 0, stream>>>(x, out);
}